// CategoricalActionHead_71150428225808
// MI455X (gfx1250) — compile-verified
//
#include <hip/hip_runtime.h>
#include <math.h>

typedef __attribute__((ext_vector_type(2))) float v2f;
typedef __attribute__((ext_vector_type(8))) float v8f;

#define DMODEL 256
#define NCHOICE 32
#define WSTRIDE 260   // W row stride in LDS floats: 260 mod 64 = 4 -> conflict-free B reads
#define LSTRIDE 33    // logits row stride in LDS floats

__global__ __launch_bounds__(256) void cat_head_kernel(
    const float* __restrict__ x_data,
    const float* __restrict__ W,
    const float* __restrict__ b,
    const int*   __restrict__ actors,
    const int*   __restrict__ mask,
    const int*   __restrict__ prev_actions,
    float* __restrict__ out_action,
    float* __restrict__ out_logprob,
    float* __restrict__ out_entropy,
    float* __restrict__ out_logp)
{
    __shared__ float w_lds[NCHOICE * WSTRIDE];     // 33280 B
    __shared__ float logits_lds[128 * LSTRIDE];    // 16896 B
    __shared__ float b_lds[NCHOICE];

    const int tid = threadIdx.x;

    // ---- stage W [32x256] into LDS with padded rows (8 x float4 per thread) ----
    #pragma unroll
    for (int j = 0; j < (NCHOICE * DMODEL) / (256 * 4); ++j) {
        const int idx4 = tid + j * 256;
        const float4 w4 = ((const float4*)W)[idx4];
        const int n = (idx4 * 4) / DMODEL;
        const int k = (idx4 * 4) % DMODEL;
        *(float4*)(&w_lds[n * WSTRIDE + k]) = w4;   // 16B aligned: WSTRIDE%4==0, k%4==0
    }
    if (tid < NCHOICE) b_lds[tid] = b[tid];
    __syncthreads();

    const int wave = tid >> 5;        // 0..7
    const int lane = tid & 31;
    const int h    = lane >> 4;       // half-wave: 0 or 1
    const int nn   = lane & 15;

    const int actor_base = blockIdx.x * 128 + wave * 16;

    // A-matrix row for actor m = nn (lanes l and l+16 split the row's K range)
    const int aidx = actors[actor_base + nn];
    const float* arow = x_data + (size_t)aidx * DMODEL + 4 * h;

    const float* wb0 = &w_lds[nn * WSTRIDE + 4 * h];          // choices 0..15
    const float* wb1 = &w_lds[(nn + 16) * WSTRIDE + 4 * h];   // choices 16..31

    v8f c0 = {};
    v8f c1 = {};

    // K permutation per 8-wide chunk c: lo step covers K={8c,8c+1,8c+4,8c+5},
    // hi step covers K={8c+2,8c+3,8c+6,8c+7}; A and B agree, sum is K-order invariant.
    #pragma unroll 4
    for (int cidx = 0; cidx < DMODEL / 8; ++cidx) {
        const float4 a4 = *(const float4*)(arow + cidx * 8);  // global_load_b128 (gather)
        const float4 b0 = *(const float4*)(wb0 + cidx * 8);   // ds_load_b128
        const float4 b1 = *(const float4*)(wb1 + cidx * 8);   // ds_load_b128
        v2f alo  = {a4.x, a4.y}, ahi  = {a4.z, a4.w};
        v2f b0lo = {b0.x, b0.y}, b0hi = {b0.z, b0.w};
        v2f b1lo = {b1.x, b1.y}, b1hi = {b1.z, b1.w};
        c0 = __builtin_amdgcn_wmma_f32_16x16x4_f32(false, alo, false, b0lo, (short)0, c0, false, false);
        c0 = __builtin_amdgcn_wmma_f32_16x16x4_f32(false, ahi, false, b0hi, (short)0, c0, false, false);
        c1 = __builtin_amdgcn_wmma_f32_16x16x4_f32(false, alo, false, b1lo, (short)0, c1, false, false);
        c1 = __builtin_amdgcn_wmma_f32_16x16x4_f32(false, ahi, false, b1hi, (short)0, c1, false, false);
    }

    // C/D layout: lane l, VGPR v -> row M = v + 8*(l>>4), col N = l&15
    #pragma unroll
    for (int v = 0; v < 8; ++v) {
        const int row = wave * 16 + v + 8 * h;
        logits_lds[row * LSTRIDE + nn]      = c0[v];
        logits_lds[row * LSTRIDE + 16 + nn] = c1[v];
    }
    __syncthreads();

    // ---- masked log-softmax epilogue: one thread per actor ----
    if (tid < 128) {
        const int a = blockIdx.x * 128 + tid;
        float* lrow = &logits_lds[tid * LSTRIDE];
        const int* mrow = mask + (size_t)a * NCHOICE;

        float mx = -INFINITY;
        #pragma unroll
        for (int c = 0; c < NCHOICE; ++c) {
            float v = lrow[c] + b_lds[c];
            if (mrow[c] == 0) v = -INFINITY;
            lrow[c] = v;
            mx = fmaxf(mx, v);
        }
        float s = 0.0f;
        #pragma unroll
        for (int c = 0; c < NCHOICE; ++c) s += __expf(lrow[c] - mx);
        const float logZ = mx + __logf(s);

        const int act = prev_actions[a];
        float lp_act = 0.0f;
        float ent = 0.0f;
        float* orow = out_logp + (size_t)a * NCHOICE;
        #pragma unroll
        for (int c = 0; c < NCHOICE; ++c) {
            const float lp = lrow[c] - logZ;         // -inf for masked entries
            orow[c] = lp;
            if (lp > -INFINITY) ent -= __expf(lp) * lp;  // 0*(-inf) guard
            if (c == act) lp_act = lp;
        }
        out_action[a]  = (float)act;
        out_logprob[a] = lp_act;
        out_entropy[a] = ent;
    }
}

extern "C" void kernel_launch(void* const* d_in, const int* in_sizes, int n_in,
                              void* d_out, int out_size, void* d_ws, size_t ws_size,
                              hipStream_t stream) {
    const float* x_data = (const float*)d_in[0];
    const float* W      = (const float*)d_in[1];
    const float* b      = (const float*)d_in[2];
    const int* actors   = (const int*)d_in[3];
    const int* mask     = (const int*)d_in[4];
    const int* prev     = (const int*)d_in[5];

    const int A = in_sizes[3];            // 262144
    float* out = (float*)d_out;
    float* out_action  = out;
    float* out_logprob = out + (size_t)A;
    float* out_entropy = out + 2 * (size_t)A;
    float* out_logp    = out + 3 * (size_t)A;

    dim3 grid(A / 128);
    cat_head_kernel<<<grid, 256, 0, stream>>>(
        x_data, W, b, actors, mask, prev,
        out_action, out_logprob, out_entropy, out_logp);
}